// MultVariateKLD_75840532512888
// MI455X (gfx1250) — compile-verified
//
#include <hip/hip_runtime.h>
#include <math.h>

// ---------------------------------------------------------------------------
// KL divergence of diagonal Gaussians built from per-row stats of m1/m2.
// Shapes: (BS=32, LEN=2048, D=512) fp32.  Memory-bound: 256 MiB @ 23.3 TB/s
// => ~11.5 us floor.  Phase 1 streams both tensors once (NT loads, 512B per
// wave instruction), computes per-row f(), reduces per block.  Phase 2 sums
// the 2048 block partials with V_WMMA_F32_16X16X4_F32 (exact fp32) and
// writes the scalar.
// ---------------------------------------------------------------------------

typedef float f4  __attribute__((ext_vector_type(4)));
typedef float v2f __attribute__((ext_vector_type(2)));
typedef float v8f __attribute__((ext_vector_type(8)));

#define D_DIM     512
#define NBLOCKS   2048
#define BLOCK_THREADS 256   // 8 wave32 waves per block

__device__ __forceinline__ float wave_sum32(float v) {
#pragma unroll
  for (int m = 16; m >= 1; m >>= 1) v += __shfl_xor(v, m, 32);
  return v;   // butterfly: all 32 lanes end with the total
}

__global__ void __launch_bounds__(BLOCK_THREADS)
kl_rows_kernel(const float* __restrict__ m1, const float* __restrict__ m2,
               float* __restrict__ partial, int rows) {
  const int lane   = threadIdx.x & 31;
  const int wid    = threadIdx.x >> 5;
  const int wpb    = BLOCK_THREADS >> 5;               // waves per block
  const int gwave  = blockIdx.x * wpb + wid;
  const int nwaves = gridDim.x * wpb;

  const float invD   = 1.0f / (float)D_DIM;
  const float invDm1 = 1.0f / (float)(D_DIM - 1);

  float acc = 0.0f;   // sum of f(row) over this wave's rows

  for (int row = gwave; row < rows; row += nwaves) {
    const f4* __restrict__ r1 = (const f4*)(m1 + (size_t)row * D_DIM);
    const f4* __restrict__ r2 = (const f4*)(m2 + (size_t)row * D_DIM);

    // Prefetch the m2 row (2 KB: 32 lanes x 64B) while m1 streams.
    __builtin_prefetch((const char*)r2 + (size_t)lane * 64, 0, 0);

    float s1 = 0.f, q1 = 0.f, s2 = 0.f, q2 = 0.f;
#pragma unroll
    for (int j = 0; j < D_DIM / (32 * 4); ++j) {       // 4 iterations
      f4 a = __builtin_nontemporal_load(&r1[lane + 32 * j]);
      s1 += (a.x + a.y) + (a.z + a.w);
      q1 += (a.x * a.x + a.y * a.y) + (a.z * a.z + a.w * a.w);
      f4 b = __builtin_nontemporal_load(&r2[lane + 32 * j]);
      s2 += (b.x + b.y) + (b.z + b.w);
      q2 += (b.x * b.x + b.y * b.y) + (b.z * b.z + b.w * b.w);
    }

    s1 = wave_sum32(s1);  q1 = wave_sum32(q1);
    s2 = wave_sum32(s2);  q2 = wave_sum32(q2);

    const float mean1 = s1 * invD;
    const float mean2 = s2 * invD;
    const float var1  = (q1 - s1 * s1 * invD) * invDm1;   // unbiased (ddof=1)
    const float var2  = (q2 - s2 * s2 * invD) * invDm1;
    const float diff  = var2 - var1;

    // f = log(mean2)-log(mean1) + mean1/mean2 + diff^2/mean2
    const float rm2 = 1.0f / mean2;
    acc += (logf(mean2) - logf(mean1)) + mean1 * rm2 + diff * diff * rm2;
  }

  __shared__ float ws[BLOCK_THREADS / 32];
  if (lane == 0) ws[wid] = acc;       // every lane holds the same total
  __syncthreads();
  if (threadIdx.x == 0) {
    float t = 0.0f;
#pragma unroll
    for (int i = 0; i < BLOCK_THREADS / 32; ++i) t += ws[i];
    partial[blockIdx.x] = t;          // deterministic: fixed order, no atomics
  }
}

// Phase 2: one wave sums NBLOCKS partials via V_WMMA_F32_16X16X4_F32.
// A-tile layout (32-bit A, 16x4, ISA 7.12.2): lanes 0-15 supply M=lane,
// K=0..1 in VGPR0..1; lanes 16-31 supply M=lane-16, K=2..3.  B = ones(4x16),
// so D[i][j] = C[i][j] + rowsum_i(A): exact fp32 accumulation of 64 partials
// per instruction.
__global__ void __launch_bounds__(32)
kl_final_kernel(const float* __restrict__ partial, float* __restrict__ out,
                float inv_bs, float n_len) {
  const int lane = threadIdx.x;          // 32 threads, EXEC all ones
  const int m    = lane & 15;
  const int kb   = (lane >> 4) * 2;

  v2f bones; bones[0] = 1.0f; bones[1] = 1.0f;
  v8f c = {};                             // zero accumulator

  for (int tile = 0; tile < NBLOCKS; tile += 64) {
    v2f a;
    a[0] = partial[tile + m * 4 + kb + 0];
    a[1] = partial[tile + m * 4 + kb + 1];
    // 8 args: (neg_a, A, neg_b, B, c_mod, C, reuse_a, reuse_b)
    c = __builtin_amdgcn_wmma_f32_16x16x4_f32(
        /*neg_a=*/false, a, /*neg_b=*/false, bones,
        /*c_mod=*/(short)0, c, /*reuse_a=*/false, /*reuse_b=*/false);
  }

  // Per C/D layout: lanes 0-15 hold rows 0..7 of their column, lanes 16-31
  // hold rows 8..15.  All columns are identical (B = ones).
  float t = ((c[0] + c[1]) + (c[2] + c[3])) + ((c[4] + c[5]) + (c[6] + c[7]));
  t += __shfl_xor(t, 16, 32);            // rows 0-7 + rows 8-15 => grand total

  if (lane == 0) out[0] = 0.5f * (t * inv_bs - n_len);
}

extern "C" void kernel_launch(void* const* d_in, const int* in_sizes, int n_in,
                              void* d_out, int out_size, void* d_ws, size_t ws_size,
                              hipStream_t stream) {
  const float* m1 = (const float*)d_in[0];
  const float* m2 = (const float*)d_in[1];
  float* out      = (float*)d_out;
  float* partials = (float*)d_ws;        // NBLOCKS floats = 8 KB scratch

  const long long total = (long long)in_sizes[0];     // BS*LEN*D
  const int rows = (int)(total / D_DIM);              // BS*LEN = 65536
  const int len  = 2048;
  const int bs   = rows / len;                        // 32

  kl_rows_kernel<<<NBLOCKS, BLOCK_THREADS, 0, stream>>>(m1, m2, partials, rows);
  kl_final_kernel<<<1, 32, 0, stream>>>(partials, out, 1.0f / (float)bs,
                                        (float)len);
}